// TP_layer_22539988369628
// MI455X (gfx1250) — compile-verified
//
#include <hip/hip_runtime.h>
#include <stdint.h>
#include <math.h>

#define BATCH      32
#define SIZE       16384
#define KNN        16
#define ND         5
#define BLOCK      1024
#define PTS_PER_WG 2048
#define CHUNKS     (SIZE / PTS_PER_WG)   // 8 chunks per batch
#define RIDGE      1e-6f

// One workgroup stages one batch's full gather table (points: 128KB, values: 64KB
// -> 192KB of the 320KB WGP LDS) via CDNA5 async global->LDS DMA, then each lane
// solves the per-point 5x5 weighted-least-squares system entirely in registers.
__launch_bounds__(BLOCK, 1)
__global__ void tp_layer_kernel(const float* __restrict__ x,
                                const float* __restrict__ points,
                                const int*   __restrict__ edge,
                                const float* __restrict__ dist,
                                const float* __restrict__ dt_p,
                                const float* __restrict__ w_p,
                                float*       __restrict__ out)
{
    __shared__ float2 s_pts[SIZE];   // 128 KB
    __shared__ float  s_val[SIZE];   //  64 KB

    const int tid   = threadIdx.x;
    const int b     = blockIdx.x / CHUNKS;
    const int chunk = blockIdx.x % CHUNKS;

    const float* pts_b = points + (size_t)b * SIZE * 2;
    const float* val_b = x      + (size_t)b * SIZE;

    // Low 32 bits of the flat address of a __shared__ object are the LDS byte
    // offset (shared aperture base has zero low bits) -> usable as async VDST.
    uint32_t lds_pts = (uint32_t)(uintptr_t)(&s_pts[0]);
    uint32_t lds_val = (uint32_t)(uintptr_t)(&s_val[0]);

    // ---- Async stage: 192KB global -> LDS (12 x b128 per lane) --------------
    // points[b]: 8192 x 16B chunks
    for (int i = tid; i < (SIZE * 2 * 4) / 16; i += BLOCK) {
        uint32_t go = (uint32_t)i * 16u;
        uint32_t lo = lds_pts + go;
        asm volatile("global_load_async_to_lds_b128 %0, %1, %2"
                     :: "v"(lo), "v"(go), "s"(pts_b) : "memory");
    }
    // x[b]: 4096 x 16B chunks
    for (int i = tid; i < (SIZE * 4) / 16; i += BLOCK) {
        uint32_t go = (uint32_t)i * 16u;
        uint32_t lo = lds_val + go;
        asm volatile("global_load_async_to_lds_b128 %0, %1, %2"
                     :: "v"(lo), "v"(go), "s"(val_b) : "memory");
    }
    asm volatile("s_wait_asynccnt 0" ::: "memory");
    __syncthreads();

    const float dt = dt_p[0];
    const float w0 = w_p[0], w1 = w_p[1], w2 = w_p[2], w3 = w_p[3], w4 = w_p[4];

    #pragma unroll
    for (int pp = 0; pp < PTS_PER_WG / BLOCK; ++pp) {
        const int p = chunk * PTS_PER_WG + pp * BLOCK + tid;

        const float2 cp = s_pts[p];
        const float  v  = s_val[p];

        // Streamed, 64B-aligned vector loads of this point's 16 indices/weights
        const size_t base = ((size_t)b * SIZE + p) * KNN;
        const int4*   e4 = (const int4*)(edge + base);
        const float4* d4 = (const float4*)(dist + base);
        int   idx[KNN];
        float dw[KNN];
        #pragma unroll
        for (int q = 0; q < 4; ++q) {
            int4   e = e4[q];
            float4 d = d4[q];
            idx[4*q+0] = e.x; idx[4*q+1] = e.y; idx[4*q+2] = e.z; idx[4*q+3] = e.w;
            dw [4*q+0] = d.x; dw [4*q+1] = d.y; dw [4*q+2] = d.z; dw [4*q+3] = d.w;
        }

        // Accumulate upper triangle of AtA (15) and Atb (5)
        float t00=0,t01=0,t02=0,t03=0,t04=0,
              t11=0,t12=0,t13=0,t14=0,
              t22=0,t23=0,t24=0,
              t33=0,t34=0,
              t44=0;
        float r0=0,r1=0,r2=0,r3=0,r4=0;

        #pragma unroll
        for (int k = 0; k < KNN; ++k) {
            const int j  = idx[k];
            const float2 np = s_pts[j];    // LDS gather (ds_load_b64)
            const float  nv = s_val[j];    // LDS gather (ds_load_b32)
            const float  wg = dw[k];
            const float dx = np.x - cp.x;
            const float dy = np.y - cp.y;
            const float a0 = dx * wg;
            const float a1 = dy * wg;
            const float a2 = 0.5f * dx * dx * wg;
            const float a3 = dx * dy * wg;
            const float a4 = 0.5f * dy * dy * wg;
            const float wr = (nv - v) * wg;
            t00 += a0*a0; t01 += a0*a1; t02 += a0*a2; t03 += a0*a3; t04 += a0*a4;
            t11 += a1*a1; t12 += a1*a2; t13 += a1*a3; t14 += a1*a4;
            t22 += a2*a2; t23 += a2*a3; t24 += a2*a4;
            t33 += a3*a3; t34 += a3*a4;
            t44 += a4*a4;
            r0 += a0*wr; r1 += a1*wr; r2 += a2*wr; r3 += a3*wr; r4 += a4*wr;
        }

        // Augmented 5x6 system, fully register-resident after unrolling
        float M[ND][ND + 1];
        M[0][0]=t00+RIDGE; M[0][1]=t01; M[0][2]=t02; M[0][3]=t03; M[0][4]=t04; M[0][5]=r0;
        M[1][0]=t01; M[1][1]=t11+RIDGE; M[1][2]=t12; M[1][3]=t13; M[1][4]=t14; M[1][5]=r1;
        M[2][0]=t02; M[2][1]=t12; M[2][2]=t22+RIDGE; M[2][3]=t23; M[2][4]=t24; M[2][5]=r2;
        M[3][0]=t03; M[3][1]=t13; M[3][2]=t23; M[3][3]=t33+RIDGE; M[3][4]=t34; M[3][5]=r3;
        M[4][0]=t04; M[4][1]=t14; M[4][2]=t24; M[4][3]=t34; M[4][4]=t44+RIDGE; M[4][5]=r4;

        // Gaussian elimination with partial pivoting (bubble max pivot up via
        // per-element selects -> pure v_cndmask, no divergence, no scratch)
        #pragma unroll
        for (int c = 0; c < ND; ++c) {
            #pragma unroll
            for (int r = c + 1; r < ND; ++r) {
                const bool sw = fabsf(M[r][c]) > fabsf(M[c][c]);
                #pragma unroll
                for (int j = c; j < ND + 1; ++j) {
                    const float hi = sw ? M[r][j] : M[c][j];
                    const float lo = sw ? M[c][j] : M[r][j];
                    M[c][j] = hi;
                    M[r][j] = lo;
                }
            }
            const float inv = 1.0f / M[c][c];
            #pragma unroll
            for (int r = c + 1; r < ND; ++r) {
                const float f = M[r][c] * inv;
                #pragma unroll
                for (int j = c + 1; j < ND + 1; ++j)
                    M[r][j] -= f * M[c][j];
            }
        }
        float sol[ND];
        #pragma unroll
        for (int c = ND - 1; c >= 0; --c) {
            float s = M[c][ND];
            #pragma unroll
            for (int j = c + 1; j < ND; ++j)
                s -= M[c][j] * sol[j];
            sol[c] = s / M[c][c];
        }

        const float du = sol[0]*w0 + sol[1]*w1 + sol[2]*w2 + sol[3]*w3 + sol[4]*w4;
        out[(size_t)b * SIZE + p] = v + dt * du;
    }
}

extern "C" void kernel_launch(void* const* d_in, const int* in_sizes, int n_in,
                              void* d_out, int out_size, void* d_ws, size_t ws_size,
                              hipStream_t stream) {
    // setup_inputs order: x, points, edge_index, dt, dist, weight
    const float* x      = (const float*)d_in[0];
    const float* points = (const float*)d_in[1];
    const int*   edge   = (const int*)  d_in[2];
    const float* dt     = (const float*)d_in[3];
    const float* dist   = (const float*)d_in[4];
    const float* weight = (const float*)d_in[5];
    float* out = (float*)d_out;

    const int batches = in_sizes[0] / SIZE;   // = 32
    dim3 grid(batches * CHUNKS);              // 256 workgroups
    dim3 block(BLOCK);                        // 1024 threads = 32 wave32 waves
    tp_layer_kernel<<<grid, block, 0, stream>>>(x, points, edge, dist, dt, weight, out);
}